// Suppression_2834678415714
// MI455X (gfx1250) — compile-verified
//
#include <hip/hip_runtime.h>

// ---------------------------------------------------------------------------
// SSD postprocess (per-class NMS + per-batch top-k) for MI455X / gfx1250.
// No matmul exists in this workload -> no WMMA. CDNA5 leverage instead:
//   * 320KB WGP LDS: whole 16K-key bitonic sort is LDS-resident (128KB).
//   * GLOBAL_LOAD_ASYNC_TO_LDS_B32 gather of the stride-25 score column.
//   * wave32 blocks (8 waves) for the <=300-wide suppression check.
// ---------------------------------------------------------------------------

#define GAS_ __attribute__((address_space(1)))
#define LAS_ __attribute__((address_space(3)))

#if defined(__HIP_DEVICE_COMPILE__) && __has_builtin(__builtin_amdgcn_global_load_async_to_lds_b32)
#define HAS_ASYNC_LDS 1
#else
#define HAS_ASYNC_LDS 0
#endif

constexpr int   kB    = 16;
constexpr int   kN    = 8732;
constexpr int   kC    = 20;
constexpr int   kRow  = 1 + kC + 4;      // 25 floats per detection row
constexpr int   kNmsK = 300;
constexpr int   kTopK = 200;
constexpr float kConfT = 0.01f;
constexpr float kIouT  = 0.45f;

constexpr int kNPad   = 16384;           // pow2 >= kN
constexpr int kRawU64 = 12016;           // u64 slot where raw-score staging starts
                                         // (bytes 96128..131055; key writes stay < 69856 -> disjoint)
constexpr int kM      = kC * kNmsK;      // 6000 candidates per batch
constexpr int kMPad   = 8192;            // pow2 >= kM

__device__ __forceinline__ unsigned orderKey(float s) {
  unsigned u = __float_as_uint(s);
  return (u & 0x80000000u) ? ~u : (u | 0x80000000u);
}
__device__ __forceinline__ float unorderKey(unsigned ou) {
  unsigned u = (ou & 0x80000000u) ? (ou & 0x7FFFFFFFu) : ~ou;
  return __uint_as_float(u);
}

// Descending bitonic sort of NP (power-of-two) 64-bit keys in LDS.
template <int NP>
__device__ __forceinline__ void bitonicDesc(unsigned long long* keys, int tid, int nt) {
  for (unsigned k = 2; k <= (unsigned)NP; k <<= 1) {
    for (unsigned j = k >> 1; j > 0; j >>= 1) {
      for (unsigned t = (unsigned)tid; t < (unsigned)NP; t += (unsigned)nt) {
        unsigned ixj = t ^ j;
        if (ixj > t) {
          unsigned long long a = keys[t], b = keys[ixj];
          bool descBlk = ((t & k) == 0);
          if (descBlk ? (a < b) : (a > b)) { keys[t] = b; keys[ixj] = a; }
        }
      }
      __syncthreads();
    }
  }
}

// One block per (batch, class): sort scores desc, greedy NMS, emit 300 slots.
__global__ __launch_bounds__(256) void nms_kernel(const float* __restrict__ in,
                                                  float* __restrict__ selS,
                                                  float* __restrict__ selB) {
  __shared__ unsigned long long keys[kNPad];                       // 128 KB
  __shared__ float sy1[kNmsK], sx1[kNmsK], sy2[kNmsK], sx2[kNmsK], ssc[kNmsK];
  __shared__ int sCnt, sFlag;

  const int tid = threadIdx.x, nt = blockDim.x;
  const int bc = blockIdx.x, b = bc / kC, c = bc % kC;
  const float* base = in + (size_t)b * kN * kRow;
  float* raw = reinterpret_cast<float*>(keys + kRawU64);           // staging tail

  // ---- Stage stride-25 score column into LDS (async gather on gfx1250) ----
#if HAS_ASYNC_LDS
  for (int t = tid; t < kN; t += nt) {
    const float* gp = base + (size_t)t * kRow + 1 + c;
    __builtin_amdgcn_global_load_async_to_lds_b32((GAS_ int*)gp, (LAS_ int*)(raw + t), 0, 0);
  }
#if __has_builtin(__builtin_amdgcn_s_wait_asynccnt)
  __builtin_amdgcn_s_wait_asynccnt(0);
#else
  asm volatile("s_wait_asynccnt 0" ::: "memory");
#endif
#else
  for (int t = tid; t < kN; t += nt) raw[t] = base[(size_t)t * kRow + 1 + c];
#endif
  __syncthreads();

  // ---- Build 64-bit keys: (ordered score bits)<<32 | (0xFFFFFFFF - idx) ----
  // Writes hit keys[0..kN), reads hit the tail slots -> disjoint, no race.
  for (int t = tid; t < kN; t += nt) {
    unsigned ou = orderKey(raw[t]);
    keys[t] = ((unsigned long long)ou << 32) |
              (unsigned long long)(0xFFFFFFFFu - (unsigned)t);
  }
  __syncthreads();
  for (int t = kN + tid; t < kNPad; t += nt) keys[t] = 0ull;       // pads sink
  for (int t = tid; t < kNmsK; t += nt) {
    ssc[t] = -1.0f;                                                // ref sentinel
    sy1[t] = sx1[t] = sy2[t] = sx2[t] = 0.0f;
  }
  if (tid == 0) sCnt = 0;
  __syncthreads();

  bitonicDesc<kNPad>(keys, tid, nt);                               // ends with barrier

  // ---- Greedy NMS: serial over sorted list, parallel 300-wide IoU check ----
  for (int i = 0; i < kN; ++i) {
    unsigned long long kk = keys[i];
    float sc = unorderKey((unsigned)(kk >> 32));
    if (!(sc > kConfT)) break;                                     // uniform
    unsigned idx = 0xFFFFFFFFu - (unsigned)(kk & 0xFFFFFFFFull);
    const float* bp = base + (size_t)idx * kRow + (1 + kC);
    float by1 = bp[0], bx1 = bp[1], by2 = bp[2], bx2 = bp[3];
    if (tid == 0 && i + 1 < kN) {                                  // prefetch next box
      unsigned nidx = 0xFFFFFFFFu - (unsigned)(keys[i + 1] & 0xFFFFFFFFull);
      if (nidx < (unsigned)kN)
        __builtin_prefetch(base + (size_t)nidx * kRow + (1 + kC), 0, 0);
    }
    const int cnt = sCnt;
    if (tid == 0) sFlag = 0;
    __syncthreads();

    bool sup = false;
    float areaA = (by2 - by1) * (bx2 - bx1);
    for (int t = tid; t < cnt; t += nt) {
      float yy1 = fmaxf(by1, sy1[t]);
      float xx1 = fmaxf(bx1, sx1[t]);
      float yy2 = fminf(by2, sy2[t]);
      float xx2 = fminf(bx2, sx2[t]);
      float inter = fmaxf(yy2 - yy1, 0.0f) * fmaxf(xx2 - xx1, 0.0f);
      float areaB = (sy2[t] - sy1[t]) * (sx2[t] - sx1[t]);
      float uni = areaA + areaB - inter;
      float iou = (uni > 0.0f) ? (inter / uni) : 0.0f;
      sup = sup || (iou > kIouT);
    }
    if (sup) atomicOr(&sFlag, 1);                                  // LDS ds_or
    __syncthreads();

    if (sFlag == 0 && cnt < kNmsK) {
      if (tid == 0) {
        sy1[cnt] = by1; sx1[cnt] = bx1; sy2[cnt] = by2; sx2[cnt] = bx2;
        ssc[cnt] = sc; sCnt = cnt + 1;
      }
    }
    __syncthreads();
    if (sCnt >= kNmsK) break;                                      // uniform
  }
  __syncthreads();

  float* oS = selS + (size_t)bc * kNmsK;
  float* oB = selB + (size_t)bc * kNmsK * 4;
  for (int t = tid; t < kNmsK; t += nt) {
    oS[t] = ssc[t];
    oB[4 * t + 0] = sy1[t]; oB[4 * t + 1] = sx1[t];
    oB[4 * t + 2] = sy2[t]; oB[4 * t + 3] = sx2[t];
  }
}

// One block per batch: top-200 of 6000 via LDS bitonic, format [cls+1, s, box].
__global__ __launch_bounds__(256) void topk_kernel(const float* __restrict__ selS,
                                                   const float* __restrict__ selB,
                                                   float* __restrict__ out) {
  __shared__ unsigned long long keys[kMPad];                       // 64 KB
  const int tid = threadIdx.x, nt = blockDim.x;
  const int b = blockIdx.x;
  const float* ss = selS + (size_t)b * kM;

  for (int t = tid; t < kMPad; t += nt) {
    unsigned long long kk = 0ull;
    if (t < kM) {
      unsigned ou = orderKey(ss[t]);
      kk = ((unsigned long long)ou << 32) |
           (unsigned long long)(0xFFFFFFFFu - (unsigned)t);
    }
    keys[t] = kk;
  }
  __syncthreads();

  bitonicDesc<kMPad>(keys, tid, nt);

  for (int k = tid; k < kTopK; k += nt) {
    unsigned long long kk = keys[k];
    float sc = unorderKey((unsigned)(kk >> 32));
    unsigned flat = 0xFFFFFFFFu - (unsigned)(kk & 0xFFFFFFFFull);
    if (flat >= (unsigned)kM) flat = 0u;                           // pad guard
    bool valid = sc > 0.0f;
    const float* bx = selB + ((size_t)b * kM + flat) * 4;
    float cls = (float)(flat / kNmsK);
    float* op = out + ((size_t)b * kTopK + k) * 6;
    op[0] = (valid ? cls : 0.0f) + 1.0f;
    op[1] = valid ? sc : 0.0f;
#pragma unroll
    for (int q = 0; q < 4; ++q)
      op[2 + q] = valid ? fminf(fmaxf(bx[q], 0.0f), 1.0f) : 0.0f;
  }
}

extern "C" void kernel_launch(void* const* d_in, const int* in_sizes, int n_in,
                              void* d_out, int out_size, void* d_ws, size_t ws_size,
                              hipStream_t stream) {
  (void)in_sizes; (void)n_in; (void)out_size; (void)ws_size;
  const float* in = (const float*)d_in[0];
  float* out = (float*)d_out;
  // Workspace: sel scores [B*C*300] then sel boxes [B*C*300*4] (~1.9 MB).
  float* selS = (float*)d_ws;
  float* selB = selS + (size_t)kB * kC * kNmsK;
  nms_kernel<<<kB * kC, 256, 0, stream>>>(in, selS, selB);
  topk_kernel<<<kB, 256, 0, stream>>>(selS, selB, out);
}